// PermutohedralRefinedUNet_13898514169859
// MI455X (gfx1250) — compile-verified
//
// PermutohedralRefinedUNet — DenseCRF mean-field on MI455X (gfx1250, wave32).
//
// Pipeline (all on `stream`, graph-capture safe):
//   1) unary/softmax init
//   2) GPU permutohedral lattice build (hash insert -> compact -> blur neighbors)
//   3) norm = filter(ones) per lattice
//   4) 10x { splat -> d+1 blur sweeps (ping-pong) -> slice } x2 lattices,
//      then compat transform via V_WMMA_F32_16X16X4_F32 + softmax.
//
// MI455X notes: irregular gather/scatter bound; lattice value arrays (<=25MB)
// sit in the 192MB L2 so blur/splat resolve at L2 rate. f32 throughout
// (float atomics); the only matmul (msg @ compat, 4x4) uses the f32 WMMA.

#include <hip/hip_runtime.h>
#include <hip/hip_bf16.h>
#include <math.h>

#define DEVFN __device__ __forceinline__

namespace {

constexpr int H = 512, W = 512, NCLS = 4;
constexpr int N = H * W;
constexpr int DBI = 5, DSP = 2;
constexpr float THETA_ALPHA = 80.0f, THETA_BETA = 0.0625f, THETA_GAMMA = 3.0f;
constexpr float BI_COMPAT = 10.0f, SP_COMPAT = 3.0f;
constexpr int N_ITER = 10;
constexpr float LOG_GT = 0.35667494393873245f;       // -ln(0.7) sign handled at use
constexpr float A_BI = 1.0f / (1.0f + 0.03125f);     // 1/(1+2^-5)
constexpr float A_SP = 1.0f / (1.0f + 0.25f);        // 1/(1+2^-2)

constexpr int NV_BI = N * (DBI + 1);   // worst-case lattice points (bilateral)
constexpr int NV_SP = N * (DSP + 1);
constexpr int CAP_BI = 1 << 22;        // hash capacity (load factor <= 0.375)
constexpr int CAP_SP = 1 << 21;

constexpr unsigned long long HT_EMPTY = ~0ull;

// ---------------- hash table (keys packed 12 bits/coord into u64) -----------

DEVFN unsigned int mix64(unsigned long long x) {
  x ^= x >> 33; x *= 0xff51afd7ed558ccdULL;
  x ^= x >> 33; x *= 0xc4ceb9fe1a85ec53ULL;
  x ^= x >> 33;
  return (unsigned int)x;
}

template <int D>
DEVFN unsigned long long pack_key(const int* k) {
  // Coordinates fit in [-2048, 2047] for both lattices (feature ranges:
  // bilateral <= ~300, spatial <= ~600). Packed value < 2^(12*D) < HT_EMPTY.
  unsigned long long r = 0;
#pragma unroll
  for (int j = 0; j < D; ++j)
    r = (r << 12) | (unsigned long long)((k[j] + 2048) & 0xFFF);
  return r;
}

template <int D>
DEVFN void unpack_key(unsigned long long kk, int* k) {
#pragma unroll
  for (int j = 0; j < D; ++j)
    k[j] = (int)((kk >> (12 * (D - 1 - j))) & 0xFFF) - 2048;
}

DEVFN int ht_insert(unsigned long long* keys, unsigned int mask, unsigned long long k) {
  unsigned int h = mix64(k) & mask;
  for (;;) {
    unsigned long long prev = atomicCAS(&keys[h], HT_EMPTY, k);
    if (prev == HT_EMPTY || prev == k) return (int)h;
    h = (h + 1) & mask;
  }
}

DEVFN int ht_find(const unsigned long long* __restrict__ keys, const int* __restrict__ idx,
                  unsigned int mask, unsigned long long k) {
  unsigned int h = mix64(k) & mask;
  for (;;) {
    unsigned long long cur = keys[h];
    if (cur == k) return idx[h];
    if (cur == HT_EMPTY) return -1;
    h = (h + 1) & mask;
  }
}

// ---------------- permutohedral embedding (port of _lattice_init) -----------

template <int D>
DEVFN void pl_embed(const float* f, float* w, int* rank, int* rem0i) {
  const float dp1 = (float)(D + 1);
  const float inv_std = sqrtf(2.0f / 3.0f) * dp1;
  float cf[D];
#pragma unroll
  for (int j = 0; j < D; ++j) {
    float sc = rsqrtf((float)((j + 2) * (j + 1))) * inv_std;
    cf[j] = f[j] * sc;
  }
  // E: row0 = ones; row i>=1: -i at col i-1, +1 at cols >= i.
  float suff[D + 1];
  suff[D] = 0.0f;
#pragma unroll
  for (int j = D - 1; j >= 0; --j) suff[j] = suff[j + 1] + cf[j];
  float elevated[D + 1];
  elevated[0] = suff[0];
#pragma unroll
  for (int i = 1; i <= D; ++i) elevated[i] = suff[i] - (float)i * cf[i - 1];

  float rem0[D + 1];
  float srem = 0.0f;
#pragma unroll
  for (int i = 0; i <= D; ++i) {
    float v = elevated[i] * (1.0f / dp1);
    float up = ceilf(v) * dp1;
    float dn = floorf(v) * dp1;
    rem0[i] = (up - elevated[i] < elevated[i] - dn) ? up : dn;
    srem += rem0[i];
  }
  int s = (int)rintf(srem * (1.0f / dp1));

#pragma unroll
  for (int i = 0; i <= D; ++i) rank[i] = 0;
#pragma unroll
  for (int i = 0; i <= D; ++i) {
    float di = elevated[i] - rem0[i];
#pragma unroll
    for (int j = i + 1; j <= D; ++j) {
      float dj = elevated[j] - rem0[j];
      if (di < dj) rank[i]++; else rank[j]++;
    }
  }
#pragma unroll
  for (int i = 0; i <= D; ++i) {
    rank[i] += s;
    if (rank[i] < 0)      { rank[i] += D + 1; rem0[i] += dp1; }
    else if (rank[i] > D) { rank[i] -= D + 1; rem0[i] -= dp1; }
  }

  float bary[D + 2];
#pragma unroll
  for (int i = 0; i < D + 2; ++i) bary[i] = 0.0f;
#pragma unroll
  for (int i = 0; i <= D; ++i) {
    float v = (elevated[i] - rem0[i]) * (1.0f / dp1);
    bary[D - rank[i]]     += v;
    bary[D - rank[i] + 1] -= v;
  }
  bary[0] += 1.0f + bary[D + 1];
#pragma unroll
  for (int i = 0; i <= D; ++i) w[i] = bary[i];
#pragma unroll
  for (int i = 0; i <= D; ++i) rem0i[i] = (int)rem0[i];
}

template <int D>
DEVFN void embed_insert(const float* f, unsigned long long* tkeys, unsigned int capMask,
                        int* off, float* wout) {
  float w[D + 1]; int rank[D + 1], rem0i[D + 1];
  pl_embed<D>(f, w, rank, rem0i);
#pragma unroll
  for (int v = 0; v <= D; ++v) {
    int key[D];
#pragma unroll
    for (int j = 0; j < D; ++j)
      key[j] = rem0i[j] + ((rank[j] < D + 1 - v) ? v : v - (D + 1));
    off[v]  = ht_insert(tkeys, capMask, pack_key<D>(key));  // slot id for now
    wout[v] = w[v];
  }
}

// ---------------- kernels ---------------------------------------------------

__global__ void k_unary(const float* __restrict__ logits,
                        float* __restrict__ unary, float* __restrict__ Q) {
  int p = blockIdx.x * blockDim.x + threadIdx.x;
  if (p >= N) return;
  float l[NCLS];
#pragma unroll
  for (int c = 0; c < NCLS; ++c) l[c] = logits[p * NCLS + c];
  float m = l[0];
#pragma unroll
  for (int c = 1; c < NCLS; ++c) m = fmaxf(m, l[c]);
  float e[NCLS], s = 0.0f;
#pragma unroll
  for (int c = 0; c < NCLS; ++c) { e[c] = __expf(l[c] - m); s += e[c]; }
  float inv = 1.0f / s, ls = __logf(s);
#pragma unroll
  for (int c = 0; c < NCLS; ++c) {
    Q[p * NCLS + c] = e[c] * inv;                      // softmax(-unary) == prob
    unary[p * NCLS + c] = (m + ls - l[c]) + LOG_GT;    // -log(prob*0.7)
  }
}

__global__ void k_embed_bi(const float* __restrict__ x, unsigned long long* __restrict__ tkeys,
                           int* __restrict__ off, float* __restrict__ wout) {
  int p = blockIdx.x * blockDim.x + threadIdx.x;
  if (p >= N) return;
  int row = p / W, col = p % W;
  float f[DBI];
  f[0] = (float)col / THETA_ALPHA;
  f[1] = (float)row / THETA_ALPHA;
  f[2] = x[p * 7 + 4] / THETA_BETA;   // image = x[..., 4:1:-1]
  f[3] = x[p * 7 + 3] / THETA_BETA;
  f[4] = x[p * 7 + 2] / THETA_BETA;
  embed_insert<DBI>(f, tkeys, CAP_BI - 1, off + p * (DBI + 1), wout + p * (DBI + 1));
}

__global__ void k_embed_sp(unsigned long long* __restrict__ tkeys,
                           int* __restrict__ off, float* __restrict__ wout) {
  int p = blockIdx.x * blockDim.x + threadIdx.x;
  if (p >= N) return;
  int row = p / W, col = p % W;
  float f[DSP] = { (float)col / THETA_GAMMA, (float)row / THETA_GAMMA };
  embed_insert<DSP>(f, tkeys, CAP_SP - 1, off + p * (DSP + 1), wout + p * (DSP + 1));
}

__global__ void k_assign(const unsigned long long* __restrict__ keys, int* __restrict__ idx,
                         int cap, int* __restrict__ counter) {
  int s = blockIdx.x * blockDim.x + threadIdx.x;
  if (s >= cap) return;
  if (keys[s] != HT_EMPTY) idx[s] = atomicAdd(counter, 1);
}

__global__ void k_finalize_off(int* __restrict__ off, const int* __restrict__ tidx, int n) {
  int i = blockIdx.x * blockDim.x + threadIdx.x;
  if (i >= n) return;
  off[i] = tidx[off[i]] + 1;  // 1-based rows; row 0 == "missing neighbor" (zeros)
}

template <int D>
__global__ void k_bn(const unsigned long long* __restrict__ keys, const int* __restrict__ idx,
                     int cap, int* __restrict__ bn) {
  int s = blockIdx.x * blockDim.x + threadIdx.x;
  if (s >= cap) return;
  unsigned long long kk = keys[s];
  if (kk == HT_EMPTY) return;
  int m = idx[s];
  int k[D];
  unpack_key<D>(kk, k);
  unsigned int mask = (unsigned int)(cap - 1);
#pragma unroll
  for (int j = 0; j <= D; ++j) {
    int n1[D], n2[D];
#pragma unroll
    for (int t = 0; t < D; ++t) {
      int bump = (t == j) ? (D + 1) : 0;
      n1[t] = k[t] - 1 + bump;
      n2[t] = k[t] + 1 - bump;
    }
    int a = ht_find(keys, idx, mask, pack_key<D>(n1));
    int b = ht_find(keys, idx, mask, pack_key<D>(n2));
    int base = (m * (D + 1) + j) * 2;
    bn[base + 0] = a + 1;   // -1 (missing) -> row 0 (always zero)
    bn[base + 1] = b + 1;
  }
}

template <int D, int C>
__global__ void k_splat(const float* __restrict__ q, const int* __restrict__ off,
                        const float* __restrict__ w, float* __restrict__ vals) {
  int p = blockIdx.x * blockDim.x + threadIdx.x;
  if (p >= N) return;
  float qv[C];
#pragma unroll
  for (int c = 0; c < C; ++c) qv[c] = (q == nullptr) ? 1.0f : q[p * C + c];
#pragma unroll
  for (int v = 0; v <= D; ++v) {
    int   o  = off[p * (D + 1) + v];
    float wt = w[p * (D + 1) + v];
#pragma unroll
    for (int c = 0; c < C; ++c) atomicAdd(&vals[o * C + c], qv[c] * wt);
  }
}

template <int C>
__global__ void k_blur(const float* __restrict__ vin, float* __restrict__ vout,
                       const int* __restrict__ bn, const int* __restrict__ Mptr,
                       int j, int nvert, int maxRows) {
  int i = blockIdx.x * blockDim.x + threadIdx.x;
  if (i >= maxRows) return;
  int M = *Mptr;
  if (i == 0) {
#pragma unroll
    for (int c = 0; c < C; ++c) vout[c] = 0.0f;   // row 0 stays zero
    return;
  }
  if (i > M) return;
  int m  = i - 1;
  int b0 = bn[(m * nvert + j) * 2 + 0];
  int b1 = bn[(m * nvert + j) * 2 + 1];
#pragma unroll
  for (int c = 0; c < C; ++c)
    vout[i * C + c] = vin[i * C + c] + 0.5f * (vin[b0 * C + c] + vin[b1 * C + c]);
}

template <int D, int C>
__global__ void k_slice(const float* __restrict__ vals, const int* __restrict__ off,
                        const float* __restrict__ w, float alpha, float* __restrict__ out) {
  int p = blockIdx.x * blockDim.x + threadIdx.x;
  if (p >= N) return;
  float acc[C];
#pragma unroll
  for (int c = 0; c < C; ++c) acc[c] = 0.0f;
#pragma unroll
  for (int v = 0; v <= D; ++v) {
    int   o  = off[p * (D + 1) + v];
    float wt = w[p * (D + 1) + v];
#pragma unroll
    for (int c = 0; c < C; ++c) acc[c] += wt * vals[o * C + c];
  }
#pragma unroll
  for (int c = 0; c < C; ++c) out[p * C + c] = alpha * acc[c];
}

// ---------------- compat transform via WMMA + softmax -----------------------
// One wave32 handles 16 pixels: A = msg tile (16x4 f32), B = compat padded to
// 4x16 (-I in the first 4 columns), D = A*B (+0) = pairwise, per ISA §7.12.2
// f32 16x16x4 lane layouts. Result staged through LDS for per-pixel softmax.

typedef float v2f_t __attribute__((ext_vector_type(2)));
typedef float v8f_t __attribute__((ext_vector_type(8)));

#if defined(__has_builtin)
#if __has_builtin(__builtin_amdgcn_wmma_f32_16x16x4_f32)
#define HAVE_WMMA_F32X4 1
#endif
#endif

__global__ void __launch_bounds__(256)
k_update(const float* __restrict__ bo, const float* __restrict__ so,
         const float* __restrict__ bi_norm, const float* __restrict__ sp_norm,
         const float* __restrict__ unary, float* __restrict__ Qout) {
  __shared__ float pw[8][16][NCLS];
  int lane = threadIdx.x & 31;
  int wv   = threadIdx.x >> 5;
  int p0   = blockIdx.x * 128 + wv * 16;
  int prow = p0 + (lane & 15);
  int kbase = (lane < 16) ? 0 : 2;      // K half held by this lane group
  int col   = lane & 15;

  float sb = BI_COMPAT / (bi_norm[prow] + 1e-20f);
  float ss = SP_COMPAT / (sp_norm[prow] + 1e-20f);

#if defined(HAVE_WMMA_F32X4)
  // A (16x4 f32): lanes 0-15 hold K=0,1 in v0,v1; lanes 16-31 hold K=2,3.
  v2f_t a;
  a.x = bo[prow * 4 + kbase + 0] * sb + so[prow * 4 + kbase + 0] * ss;
  a.y = bo[prow * 4 + kbase + 1] * sb + so[prow * 4 + kbase + 1] * ss;
  // B (4x16 f32): lanes 0-15 -> rows K=0,1; lanes 16-31 -> rows K=2,3.
  v2f_t b;
  b.x = (col == kbase + 0) ? -1.0f : 0.0f;
  b.y = (col == kbase + 1) ? -1.0f : 0.0f;
  v8f_t c = {};
  v8f_t d = __builtin_amdgcn_wmma_f32_16x16x4_f32(
      /*neg_a=*/false, a, /*neg_b=*/false, b,
      /*c_mod=*/(short)0, c, /*reuse_a=*/false, /*reuse_b=*/false);
  // D (16x16 f32): VGPR r -> row r (lanes 0-15, N=lane) / row r+8 (lanes 16-31).
  int rbase = (lane < 16) ? 0 : 8;
  if (col < NCLS) {
#pragma unroll
    for (int r = 0; r < 8; ++r) pw[wv][rbase + r][col] = d[r];
  }
#else
  if (lane < 16) {
#pragma unroll
    for (int cc = 0; cc < NCLS; ++cc) {
      float msg = bo[prow * 4 + cc] * sb + so[prow * 4 + cc] * ss;
      pw[wv][lane][cc] = -msg;   // compat = -I
    }
  }
#endif
  __syncthreads();
  if (lane < 16) {
    int pix = p0 + lane;
    float t[NCLS];
    float m = -3.4e38f;
#pragma unroll
    for (int cc = 0; cc < NCLS; ++cc) {
      t[cc] = -unary[pix * 4 + cc] - pw[wv][lane][cc];
      m = fmaxf(m, t[cc]);
    }
    float e[NCLS], s = 0.0f;
#pragma unroll
    for (int cc = 0; cc < NCLS; ++cc) { e[cc] = __expf(t[cc] - m); s += e[cc]; }
    float inv = 1.0f / s;
#pragma unroll
    for (int cc = 0; cc < NCLS; ++cc) Qout[pix * 4 + cc] = e[cc] * inv;
  }
}

// ---------------- host-side filter driver -----------------------------------

template <int D, int C>
static void run_pl_filter(const float* q, const int* off, const float* w,
                          const int* bn, const int* Mptr, int nvmax, float alpha,
                          float* valsA, float* valsB, float* out, hipStream_t stream) {
  const int TB = 256;
  int rows = nvmax + 2;
  hipMemsetAsync(valsA, 0, (size_t)rows * C * sizeof(float), stream);
  k_splat<D, C><<<(N + TB - 1) / TB, TB, 0, stream>>>(q, off, w, valsA);
  const float* vin = valsA;
  float* vout = valsB;
  for (int j = 0; j <= D; ++j) {
    k_blur<C><<<(rows + TB - 1) / TB, TB, 0, stream>>>(vin, vout, bn, Mptr, j, D + 1, rows);
    const float* t = vout; vout = const_cast<float*>(vin); vin = t;
  }
  k_slice<D, C><<<(N + TB - 1) / TB, TB, 0, stream>>>(vin, off, w, alpha, out);
}

} // namespace

extern "C" void kernel_launch(void* const* d_in, const int* in_sizes, int n_in,
                              void* d_out, int out_size, void* d_ws, size_t ws_size,
                              hipStream_t stream) {
  (void)in_sizes; (void)n_in; (void)out_size; (void)ws_size;
  const float* x      = (const float*)d_in[0];   // (H,W,7) f32
  const float* logits = (const float*)d_in[1];   // (H,W,4) f32
  float* out          = (float*)d_out;           // (H,W,4) f32

  // Workspace carve-out (~281 MB worst case).
  char* p = (char*)d_ws;
  auto take = [&](size_t bytes) -> char* {
    char* r = p; p += (bytes + 255) & ~(size_t)255; return r;
  };
  unsigned long long* bi_keys = (unsigned long long*)take((size_t)CAP_BI * 8);
  int*   bi_idx   = (int*)  take((size_t)CAP_BI * 4);
  unsigned long long* sp_keys = (unsigned long long*)take((size_t)CAP_SP * 8);
  int*   sp_idx   = (int*)  take((size_t)CAP_SP * 4);
  int*   bi_off   = (int*)  take((size_t)NV_BI * 4);
  float* bi_w     = (float*)take((size_t)NV_BI * 4);
  int*   sp_off   = (int*)  take((size_t)NV_SP * 4);
  float* sp_w     = (float*)take((size_t)NV_SP * 4);
  int*   bi_bn    = (int*)  take((size_t)NV_BI * (DBI + 1) * 2 * 4);
  int*   sp_bn    = (int*)  take((size_t)NV_SP * (DSP + 1) * 2 * 4);
  float* bi_valsA = (float*)take((size_t)(NV_BI + 2) * NCLS * 4);
  float* bi_valsB = (float*)take((size_t)(NV_BI + 2) * NCLS * 4);
  float* sp_valsA = (float*)take((size_t)(NV_SP + 2) * NCLS * 4);
  float* sp_valsB = (float*)take((size_t)(NV_SP + 2) * NCLS * 4);
  float* bi_norm  = (float*)take((size_t)N * 4);
  float* sp_norm  = (float*)take((size_t)N * 4);
  float* unary    = (float*)take((size_t)N * NCLS * 4);
  float* Q        = (float*)take((size_t)N * NCLS * 4);
  float* bo       = (float*)take((size_t)N * NCLS * 4);
  float* so       = (float*)take((size_t)N * NCLS * 4);
  int*   counters = (int*)  take(256);   // [0]=M_bi, [1]=M_sp

  const int TB = 256;
  const int gN = (N + TB - 1) / TB;

  // Init hash tables (all-ones == HT_EMPTY) + counters.
  hipMemsetAsync(bi_keys, 0xFF, (size_t)CAP_BI * 8, stream);
  hipMemsetAsync(sp_keys, 0xFF, (size_t)CAP_SP * 8, stream);
  hipMemsetAsync(counters, 0, 256, stream);

  // Unary + Q0.
  k_unary<<<gN, TB, 0, stream>>>(logits, unary, Q);

  // Lattice build.
  k_embed_bi<<<gN, TB, 0, stream>>>(x, bi_keys, bi_off, bi_w);
  k_embed_sp<<<gN, TB, 0, stream>>>(sp_keys, sp_off, sp_w);
  k_assign<<<(CAP_BI + TB - 1) / TB, TB, 0, stream>>>(bi_keys, bi_idx, CAP_BI, &counters[0]);
  k_assign<<<(CAP_SP + TB - 1) / TB, TB, 0, stream>>>(sp_keys, sp_idx, CAP_SP, &counters[1]);
  k_finalize_off<<<(NV_BI + TB - 1) / TB, TB, 0, stream>>>(bi_off, bi_idx, NV_BI);
  k_finalize_off<<<(NV_SP + TB - 1) / TB, TB, 0, stream>>>(sp_off, sp_idx, NV_SP);
  k_bn<DBI><<<(CAP_BI + TB - 1) / TB, TB, 0, stream>>>(bi_keys, bi_idx, CAP_BI, bi_bn);
  k_bn<DSP><<<(CAP_SP + TB - 1) / TB, TB, 0, stream>>>(sp_keys, sp_idx, CAP_SP, sp_bn);

  // Norms: filter(ones), C=1 (reuses the C=4 value buffers).
  run_pl_filter<DBI, 1>(nullptr, bi_off, bi_w, bi_bn, &counters[0], NV_BI, A_BI,
                        bi_valsA, bi_valsB, bi_norm, stream);
  run_pl_filter<DSP, 1>(nullptr, sp_off, sp_w, sp_bn, &counters[1], NV_SP, A_SP,
                        sp_valsA, sp_valsB, sp_norm, stream);

  // Mean-field iterations.
  for (int it = 0; it < N_ITER; ++it) {
    run_pl_filter<DBI, NCLS>(Q, bi_off, bi_w, bi_bn, &counters[0], NV_BI, A_BI,
                             bi_valsA, bi_valsB, bo, stream);
    run_pl_filter<DSP, NCLS>(Q, sp_off, sp_w, sp_bn, &counters[1], NV_SP, A_SP,
                             sp_valsA, sp_valsB, so, stream);
    float* qdst = (it == N_ITER - 1) ? out : Q;
    k_update<<<N / 128, 256, 0, stream>>>(bo, so, bi_norm, sp_norm, unary, qdst);
  }
}